// FusedSparseLMHead_71494025609807
// MI455X (gfx1250) — compile-verified
//
#include <hip/hip_runtime.h>
#include <hip/hip_bf16.h>
#include <math.h>

// ---------------------------------------------------------------------------
// Fused sparse LM-head weighted cross-entropy (chunked-vocab online LSE)
//   h[N=4096, D=2048] f32, W[V=65536, D=2048] f32, labels[N] i32, lw[N] f32
//
// GEMM: D[M=vocab, N=token] = W_tile x h_tile^T using V_WMMA_F32_16X16X4_F32.
// Grid: x = token blocks (256 tokens), y = vocab slices (4096 rows).
// Block: 8 waves. Wave w owns 32 tokens (2 x 16-token subtiles) and covers
// all 128 vocab rows of the tile -> 8x2 C tiles (128 accum VGPRs).
// Staging uses GLOBAL_LOAD_ASYNC_TO_LDS_B128 (ASYNCcnt-tracked DMA, no VGPR
// transit): issue async for K-tile kt+1, WMMA kt from LDS, s_wait_asynccnt,
// one barrier per K-tile.
// ---------------------------------------------------------------------------

typedef float v2f __attribute__((ext_vector_type(2)));
typedef float v8f __attribute__((ext_vector_type(8)));

#define N_TOK      4096
#define DIM        2048
#define VOCAB      65536
#define NSLICE     16
#define SLICE_ROWS (VOCAB / NSLICE)   // 4096
#define TM         256                 // tokens per block (2 subtiles / wave)
#define TV         128                 // vocab rows per tile
#define KB         16                  // K tile (floats)
#define LDS_STRIDE (KB + 4)            // 20: pad to dodge bank conflicts
#define NKT        (DIM / KB)          // 128 K tiles
#define IGNORE_INDEX (-100)

// async DMA: global memory -> LDS, 16 bytes per lane, tracked by ASYNCcnt
__device__ __forceinline__ void async_ld_b128(unsigned lds_off, const void* gptr) {
    asm volatile("global_load_async_to_lds_b128 %0, %1, off"
                 :: "v"(lds_off), "v"(gptr) : "memory");
}
__device__ __forceinline__ void wait_asynccnt0() {
    asm volatile("s_wait_asynccnt 0x0" ::: "memory");
}

__global__ __launch_bounds__(256)
void lmhead_gemm_lse_kernel(const float* __restrict__ h,
                            const float* __restrict__ W,
                            const int*   __restrict__ labels,
                            float* __restrict__ part_max,   // [NSLICE][N_TOK]
                            float* __restrict__ part_sum,   // [NSLICE][N_TOK]
                            float* __restrict__ lab_ws)     // [N_TOK]
{
    // double-buffered staging: 2*(256*20 + 128*20)*4 = 61440 B
    __shared__ __align__(16) float Hs[2][TM][LDS_STRIDE];
    __shared__ __align__(16) float Ws[2][TV][LDS_STRIDE];

    const int tid   = threadIdx.x;
    const int wave  = tid >> 5;
    const int lane  = tid & 31;
    const int laneN = lane & 15;
    const int half  = lane >> 4;            // 0: M 0..7, 1: M 8..15 of C tile

    const int tblock = blockIdx.x * TM;
    const int slice0 = blockIdx.y * SLICE_ROWS;

    const int hrow0  = wave * 32 + laneN;            // token subtile 0 row
    const int hrow1  = hrow0 + 16;                   // token subtile 1 row
    const int token0 = tblock + hrow0;
    const int token1 = tblock + hrow1;

    const int lab0  = labels[token0];
    const int lab1  = labels[token1];
    const int safe0 = lab0 < 0 ? 0 : lab0;
    const int safe1 = lab1 < 0 ? 0 : lab1;

    // per-thread staging slots: thread covers rows lin>>4, cols lin&15
    // Hs: 4 x b128 (4096 floats), Ws: 2 x b128 (2048 floats)
    int hrowp[4], hcolp[4], wrowp[2], wcolp[2];
    unsigned lds_h[2][4], lds_w[2][2];
    #pragma unroll
    for (int p = 0; p < 4; ++p) {
        int lin = p * 1024 + tid * 4;
        hrowp[p] = lin >> 4;  hcolp[p] = lin & 15;
        lds_h[0][p] = (unsigned)(uintptr_t)(void*)&Hs[0][hrowp[p]][hcolp[p]];
        lds_h[1][p] = (unsigned)(uintptr_t)(void*)&Hs[1][hrowp[p]][hcolp[p]];
    }
    #pragma unroll
    for (int p = 0; p < 2; ++p) {
        int lin = p * 1024 + tid * 4;
        wrowp[p] = lin >> 4;  wcolp[p] = lin & 15;
        lds_w[0][p] = (unsigned)(uintptr_t)(void*)&Ws[0][wrowp[p]][wcolp[p]];
        lds_w[1][p] = (unsigned)(uintptr_t)(void*)&Ws[1][wrowp[p]][wcolp[p]];
    }

    float rmax[2] = {-INFINITY, -INFINITY};
    float rsum[2] = {0.0f, 0.0f};

    for (int vt = 0; vt < SLICE_ROWS; vt += TV) {
        const int vtile = slice0 + vt;

        v8f acc[8][2];
        #pragma unroll
        for (int su = 0; su < 8; ++su) {
            acc[su][0] = (v8f){0,0,0,0,0,0,0,0};
            acc[su][1] = (v8f){0,0,0,0,0,0,0,0};
        }

        // -------- prologue: async-stage K-tile 0 into LDS buf 0 ----------
        #pragma unroll
        for (int p = 0; p < 4; ++p)
            async_ld_b128(lds_h[0][p],
                          &h[(size_t)(tblock + hrowp[p]) * DIM + hcolp[p]]);
        #pragma unroll
        for (int p = 0; p < 2; ++p)
            async_ld_b128(lds_w[0][p],
                          &W[(size_t)(vtile + wrowp[p]) * DIM + wcolp[p]]);
        wait_asynccnt0();
        __syncthreads();

        for (int kt = 0; kt < NKT; ++kt) {
            const int cur = kt & 1;

            // ---- issue async DMA for K-tile kt+1 into the other buffer --
            if (kt + 1 < NKT) {
                const int k0n = (kt + 1) * KB;
                const int nb  = cur ^ 1;
                #pragma unroll
                for (int p = 0; p < 4; ++p)
                    async_ld_b128(lds_h[nb][p],
                        &h[(size_t)(tblock + hrowp[p]) * DIM + k0n + hcolp[p]]);
                #pragma unroll
                for (int p = 0; p < 2; ++p)
                    async_ld_b128(lds_w[nb][p],
                        &W[(size_t)(vtile + wrowp[p]) * DIM + k0n + wcolp[p]]);
            }

            // ---- WMMA f32 16x16x4 over K-tile kt from LDS buffer cur ----------
            #pragma unroll
            for (int ks = 0; ks < KB; ks += 4) {
                const int col = ks + 2 * half;
                v2f b0 = *(const v2f*)&Hs[cur][hrow0][col];
                v2f b1 = *(const v2f*)&Hs[cur][hrow1][col];
                #pragma unroll
                for (int su = 0; su < 8; ++su) {
                    v2f a = *(const v2f*)&Ws[cur][su * 16 + laneN][col];
                    acc[su][0] = __builtin_amdgcn_wmma_f32_16x16x4_f32(
                        false, a, false, b0, (short)0, acc[su][0], false, false);
                    acc[su][1] = __builtin_amdgcn_wmma_f32_16x16x4_f32(
                        false, a, false, b1, (short)0, acc[su][1], false, false);
                }
            }

            wait_asynccnt0();     // DMA for kt+1 has landed in LDS
            __syncthreads();
        }

        // -------- label-logit capture (raw logit, single writer) ---------
        #pragma unroll
        for (int tt = 0; tt < 2; ++tt) {
            const int safe = tt ? safe1 : safe0;
            if (safe >= vtile && safe < vtile + TV) {
                const int lr = safe - vtile;
                const int su_t = lr >> 4;
                const int m    = lr & 15;
                const bool pick = half ? (m >= 8) : (m < 8);
                if (pick) {
                    const int r = m & 7;
                    float val = 0.0f;
                    #pragma unroll
                    for (int s = 0; s < 8; ++s)
                        #pragma unroll
                        for (int q = 0; q < 8; ++q)
                            if (s == su_t && q == r) val = acc[s][tt][q];
                    lab_ws[tt ? token1 : token0] = val;
                }
            }
        }

        // -------- online (max, sumexp) update over this 128-row tile -----
        #pragma unroll
        for (int tt = 0; tt < 2; ++tt) {
            float tmax = -INFINITY;
            #pragma unroll
            for (int s = 0; s < 8; ++s)
                #pragma unroll
                for (int q = 0; q < 8; ++q)
                    tmax = fmaxf(tmax, acc[s][tt][q]);
            tmax = fmaxf(tmax, __shfl_xor(tmax, 16));   // merge lane halves

            const float nmax = fmaxf(rmax[tt], tmax);
            float ssum = 0.0f;
            #pragma unroll
            for (int s = 0; s < 8; ++s)
                #pragma unroll
                for (int q = 0; q < 8; ++q)
                    ssum += __expf(acc[s][tt][q] - nmax);
            ssum += __shfl_xor(ssum, 16);               // merge lane halves

            rsum[tt] = rsum[tt] * __expf(rmax[tt] - nmax) + ssum;
            rmax[tt] = nmax;
        }
    }

    // both lane halves hold identical per-token state; half 0 writes
    if (half == 0) {
        part_max[blockIdx.y * N_TOK + token0] = rmax[0];
        part_sum[blockIdx.y * N_TOK + token0] = rsum[0];
        part_max[blockIdx.y * N_TOK + token1] = rmax[1];
        part_sum[blockIdx.y * N_TOK + token1] = rsum[1];
    }
}

// ---------------------------------------------------------------------------
// Stage 2: merge slice partials per token, per-block deterministic reduction.
// ---------------------------------------------------------------------------
__global__ __launch_bounds__(256)
void lmhead_finalize_kernel(const int*   __restrict__ labels,
                            const float* __restrict__ lw,
                            const float* __restrict__ part_max,
                            const float* __restrict__ part_sum,
                            const float* __restrict__ lab_ws,
                            float* __restrict__ blk_t,   // [gridDim.x]
                            float* __restrict__ blk_w)   // [gridDim.x]
{
    const int t = blockIdx.x * blockDim.x + threadIdx.x;

    float m = -INFINITY;
    #pragma unroll
    for (int y = 0; y < NSLICE; ++y)
        m = fmaxf(m, part_max[y * N_TOK + t]);
    float s = 0.0f;
    #pragma unroll
    for (int y = 0; y < NSLICE; ++y)
        s += __expf(part_max[y * N_TOK + t] - m) * part_sum[y * N_TOK + t];

    const float lse = m + __logf(s);
    const int lab   = labels[t];
    const float w   = (lab != IGNORE_INDEX) ? lw[t] : 0.0f;
    float term      = w * (lse - lab_ws[t]);
    float wsum      = w;

    // wave32 tree reduction then cross-wave via LDS
    #pragma unroll
    for (int o = 16; o > 0; o >>= 1) {
        term += __shfl_down(term, o);
        wsum += __shfl_down(wsum, o);
    }
    __shared__ float st[8], sw[8];
    const int wv = threadIdx.x >> 5;
    if ((threadIdx.x & 31) == 0) { st[wv] = term; sw[wv] = wsum; }
    __syncthreads();
    if (threadIdx.x == 0) {
        float a = 0.0f, b = 0.0f;
        #pragma unroll
        for (int i = 0; i < 8; ++i) { a += st[i]; b += sw[i]; }
        blk_t[blockIdx.x] = a;
        blk_w[blockIdx.x] = b;
    }
}

__global__ void lmhead_combine_kernel(const float* __restrict__ blk_t,
                                      const float* __restrict__ blk_w,
                                      int nblk, float* __restrict__ out)
{
    float a = 0.0f, b = 0.0f;
    for (int i = 0; i < nblk; ++i) { a += blk_t[i]; b += blk_w[i]; }
    out[0] = a / fmaxf(b, 1e-8f);
}

// ---------------------------------------------------------------------------
extern "C" void kernel_launch(void* const* d_in, const int* in_sizes, int n_in,
                              void* d_out, int out_size, void* d_ws, size_t ws_size,
                              hipStream_t stream)
{
    const float* h      = (const float*)d_in[0];
    const float* W      = (const float*)d_in[1];
    const int*   labels = (const int*)d_in[2];
    const float* lw     = (const float*)d_in[3];
    float* out = (float*)d_out;

    // workspace layout (floats)
    float* part_max = (float*)d_ws;                       // NSLICE * N_TOK
    float* part_sum = part_max + NSLICE * N_TOK;          // NSLICE * N_TOK
    float* lab_ws   = part_sum + NSLICE * N_TOK;          // N_TOK
    float* blk_t    = lab_ws + N_TOK;                     // 16
    float* blk_w    = blk_t + 16;                         // 16

    dim3 grid(N_TOK / TM, NSLICE);   // 16 x 16
    lmhead_gemm_lse_kernel<<<grid, 256, 0, stream>>>(
        h, W, labels, part_max, part_sum, lab_ws);

    lmhead_finalize_kernel<<<N_TOK / 256, 256, 0, stream>>>(
        labels, lw, part_max, part_sum, lab_ws, blk_t, blk_w);

    lmhead_combine_kernel<<<1, 1, 0, stream>>>(blk_t, blk_w, N_TOK / 256, out);
}